// AdSBHNet_88347477279059
// MI455X (gfx1250) — compile-verified
//
#include <hip/hip_runtime.h>
#include <math.h>

#define MQ   1000
#define PI_F 3.14159265358979323846f
#define EPSF 1e-12f
#define LN2F 0.69314718055994531f

typedef float v2f __attribute__((ext_vector_type(2)));
typedef float v8f __attribute__((ext_vector_type(8)));

// y-grid: linspace(0.001, 0.999, 1000); y2-grid: linspace(0.001, 1.0, 1000)
#define Y0F 0.001f
#define Y1F 0.999f
#define HF  (0.998f / 999.0f)
#define H2F (0.999f / 999.0f)

// Native 1-ulp hardware ops (v_rcp_f32 / v_rsq_f32 / v_sqrt_f32 / v_log_f32)
__device__ __forceinline__ float frcp (float x) { return __builtin_amdgcn_rcpf(x);  }
__device__ __forceinline__ float frsq (float x) { return __builtin_amdgcn_rsqf(x);  }
__device__ __forceinline__ float fsqrt(float x) { return __builtin_amdgcn_sqrtf(x); }
__device__ __forceinline__ float flog (float x) { return __builtin_amdgcn_logf(x) * LN2F; }

// ---------------------------------------------------------------------------
// Wave-wide sum via V_WMMA_F32_16X16X4_F32 (A: lane value in v0, 0 in v1;
// B = all-ones, layout-invariant).  D[m][n] = rowsum_m; summing the 8 C/D
// VGPRs gives rows 0-7 (lanes 0-15) / 8-15 (lanes 16-31); xor-16 combines.
// Requires EXEC = all ones (callers keep wave-uniform control flow).
// ---------------------------------------------------------------------------
__device__ __forceinline__ float wave_sum(float x) {
    v2f A; A.x = x;    A.y = 0.0f;
    v2f B; B.x = 1.0f; B.y = 1.0f;
    v8f C = {};
    C = __builtin_amdgcn_wmma_f32_16x16x4_f32(false, A, false, B,
                                              (short)0, C, false, false);
    float s = C[0] + C[1] + C[2] + C[3] + C[4] + C[5] + C[6] + C[7];
    s += __shfl_xor(s, 16, 32);
    return s;
}

// ---------------------------------------------------------------------------
// Model polynomials (s1=2a0, s2=a0^2+2a1, s3=2a0a1, s4=a1^2;
// P = 1 + b0 z + b1 z^2 + (a0+a1-b0-b1) z^3; g = P^2/(1-z^4);
// z*dg/g = 2 z P'/P + 4 z^4/(1-z^4))
// ---------------------------------------------------------------------------
struct Params { float s1, s2, s3, s4, c1, c2, c3; };

__device__ __forceinline__ Params make_params(const float* a, const float* b) {
    Params p;
    float a0 = a[0], a1 = a[1], b0 = b[0], b1 = b[1];
    p.s1 = 2.0f * a0;
    p.s2 = a0 * a0 + 2.0f * a1;
    p.s3 = 2.0f * a0 * a1;
    p.s4 = a1 * a1;
    p.c1 = b0; p.c2 = b1; p.c3 = a0 + a1 - b0 - b1;
    return p;
}

__device__ __forceinline__ float eval_f(const Params& p, float z) {
    float z2 = z * z, z3 = z2 * z, z4 = z2 * z2;
    return (1.0f - z4)
         + 1.33333333333333333f * p.s1 * (z  - z4)
         + 2.0f * p.s2 * (z2 - z4)
         + 4.0f * p.s3 * (z3 - z4)
         - 4.0f * p.s4 * z4 * flog(z);
}

__device__ __forceinline__ float eval_df(const Params& p, float z, float f) {
    float z2 = z * z, z3 = z2 * z, z4 = z2 * z2;
    float poly = 1.0f + p.s1 * z + p.s2 * z2 + p.s3 * z3 + p.s4 * z4;
    return 4.0f * (f - poly) * frcp(z);
}

// trapz over extended grid [0, y0..y999, 1], ii = [v0, e0..e999, 0]
__device__ __forceinline__ float trapz_ext(float S, float e0, float e1, float eN) {
    float v0 = e0 - (e1 - e0) * (Y0F / HF);
    return HF * (S - 0.5f * (e0 + eN))
         + 0.5f * (v0 + e0) * Y0F
         + 0.5f * eN * (1.0f - Y1F);
}

// ---------------------------------------------------------------------------
// Per-point integrands (shared by wave and block reductions)
// ---------------------------------------------------------------------------
__device__ __forceinline__ void point_LdL(const Params& p, float zs, float rfs,
                                          float Ac, int m, float& eL, float& edL) {
    float y  = Y0F + HF * (float)m;
    float u  = (1.0f - y) * (1.0f + y);                // 1 - y^2
    float z  = zs * u;
    float z2 = z * z, z3 = z2 * z, z4 = z2 * z2;
    float lg = flog(z);
    float f  = (1.0f - z4)
             + 1.33333333333333333f * p.s1 * (z  - z4)
             + 2.0f * p.s2 * (z2 - z4)
             + 4.0f * p.s3 * (z3 - z4)
             - 4.0f * p.s4 * z4 * lg;
    float P  = 1.0f + z * (p.c1 + z * (p.c2 + z * p.c3));
    float Pp = p.c1 + z * (2.0f * p.c2 + z * (3.0f * p.c3));
    float rD = frcp(1.0f - z4);
    float sqrtg = fsqrt(P * P * rD);
    float ru = frcp(u), ru2 = ru * ru, ru3 = ru2 * ru, r4 = ru2 * ru2;
    float fofs = f * rfs;
    float q  = fmaxf(fofs * r4 - 1.0f, EPSF);          // == arg_L
    float rq = frsq(q);
    eL = sqrtg * rq * y;
    float poly = 1.0f + p.s1 * z + p.s2 * z2 + p.s3 * z3 + p.s4 * z4;
    float df = 4.0f * (f - poly) * frcp(z);
    float t  = 2.0f * z * Pp * frcp(P) + 4.0f * z4 * rD;   // z*dg/g
    float integ = r4 * fofs * (Ac + t) - zs * ru3 * df * rfs - 2.0f - t;
    edL = integ * 2.0f * fsqrt(fmaxf(1.0f - u, 0.0f))
        * sqrtg * frcp(q) * rq;                        // / q^1.5
}

__device__ __forceinline__ float point_Vc(const Params& p, float zs, float fs, int m) {
    float y = Y0F + HF * (float)m;
    float u = (1.0f - y) * (1.0f + y);
    float z = zs * u;
    float z2 = z * z, z4 = z2 * z2;
    float f = eval_f(p, z);
    float P = 1.0f + z * (p.c1 + z * (p.c2 + z * p.c3));
    float g = P * P * frcp(1.0f - z4);
    float w = u * u;
    float arg = 1.0f - w * w * fs * frcp(f);           // 1 - w^2/(f/fs)
    float r = frsq(fmaxf(arg, EPSF)) - 1.0f;
    return fsqrt(fmaxf(f * g, EPSF)) * frcp(w) * r * y;
}

__device__ __forceinline__ float point_Vd(const Params& p, float zs, int m) {
    float y2 = Y0F + H2F * (float)m;
    float zv = 1.0f - (1.0f - zs) * y2;
    float zq = zv * zv;
    float f = eval_f(p, zv);
    float P = 1.0f + zv * (p.c1 + zv * (p.c2 + zv * p.c3));
    float g = P * P * frcp(1.0f - zq * zq);
    float rz = frcp(zv);
    return fsqrt(fmaxf(f * g, EPSF)) * rz * rz;
}

// ---------------------------------------------------------------------------
// Wave-level fused integrate_L + integrate_dL.  Endpoints m=0,1 live on lanes
// 0,1 (first iteration) and m=999 on lane 7 (last iteration) -> shuffles
// instead of redundant re-evaluation.
// ---------------------------------------------------------------------------
__device__ void wave_L_dL(const Params& p, float zs, int lane,
                          float& Lout, float& dLout) {
    float fs  = eval_f(p, zs), rfs = frcp(fs);
    float dfs = eval_df(p, zs, fs);
    float Ac  = zs * dfs * rfs + 2.0f;

    float SL = 0.0f, SdL = 0.0f;
    float fL = 0.0f, fdL = 0.0f, lL = 0.0f, ldL = 0.0f;
    for (int m = lane; m < MQ; m += 32) {
        float eL, edL;
        point_LdL(p, zs, rfs, Ac, m, eL, edL);
        if (m < 32) { fL = eL; fdL = edL; }            // first iteration
        lL = eL; ldL = edL;                            // last iteration
        SL += eL; SdL += edL;
    }
    SL  = wave_sum(SL);
    SdL = wave_sum(SdL);
    float e0L = __shfl(fL, 0, 32), e1L = __shfl(fL, 1, 32), eNL = __shfl(lL, 7, 32);
    float e0d = __shfl(fdL, 0, 32), e1d = __shfl(fdL, 1, 32), eNd = __shfl(ldL, 7, 32);
    Lout  = 4.0f * zs * trapz_ext(SL, e0L, e1L, eNL) * (1.0f / PI_F);
    dLout =             trapz_ext(SdL, e0d, e1d, eNd) * (1.0f / PI_F);
}

// Wave-level integrate_V (connected + disconnected fused)
__device__ float wave_V(const Params& p, float zs, float coef, int lane) {
    float fs = eval_f(p, zs);
    float Sc = 0.0f, Sd = 0.0f;
    float fc = 0.0f, lcv = 0.0f, fd = 0.0f, ldv = 0.0f;
    for (int m = lane; m < MQ; m += 32) {
        float ec = point_Vc(p, zs, fs, m);
        float ed = point_Vd(p, zs, m);
        if (m < 32) { fc = ec; fd = ed; }
        lcv = ec; ldv = ed;
        Sc += ec; Sd += ed;
    }
    Sc = wave_sum(Sc);
    Sd = wave_sum(Sd);
    float e0c = __shfl(fc, 0, 32), e1c = __shfl(fc, 1, 32), eNc = __shfl(lcv, 7, 32);
    float e0d = __shfl(fd, 0, 32), eNd = __shfl(ldv, 7, 32);
    float Tc = trapz_ext(Sc, e0c, e1c, eNc);
    float Vc = coef * PI_F * 4.0f * Tc * frcp(zs);
    float Td = 0.5f * (1.0f + e0d) * Y0F + H2F * (Sd - 0.5f * (e0d + eNd));
    return Vc - coef * PI_F * 2.0f * (1.0f - zs) * Td;
}

// ---------------------------------------------------------------------------
// Kernel 1: ONE 1024-thread block (32 waves), one quadrature point per thread,
// per-wave WMMA reduction + LDS block reduction.  Serial latency of the
// 60+ bisection integrals drops ~10x vs a single wave.
// ws[0..3] = {zs_max, L_max, zs_crit, L_crit}
// ---------------------------------------------------------------------------
__global__ __launch_bounds__(1024)
void adsbh_precompute(const float* __restrict__ a, const float* __restrict__ b,
                      const float* __restrict__ logcoef, float* __restrict__ ws) {
    __shared__ float redA[32], redB[32], ends[8];
    int tid = threadIdx.x, lane = tid & 31, wid = tid >> 5;
    Params p = make_params(a, b);

    auto block_LdL = [&](float zs, float& Lout, float& dLout) {
        float fs  = eval_f(p, zs), rfs = frcp(fs);
        float dfs = eval_df(p, zs, fs);
        float Ac  = zs * dfs * rfs + 2.0f;
        float eL = 0.0f, edL = 0.0f;
        if (tid < MQ) point_LdL(p, zs, rfs, Ac, tid, eL, edL);
        if      (tid == 0)      { ends[0] = eL; ends[1] = edL; }
        else if (tid == 1)      { ends[2] = eL; ends[3] = edL; }
        else if (tid == MQ - 1) { ends[4] = eL; ends[5] = edL; }
        float sL = wave_sum(eL), sdL = wave_sum(edL);  // uniform flow, full EXEC
        if (lane == 0) { redA[wid] = sL; redB[wid] = sdL; }
        __syncthreads();
        float SL = 0.0f, SdL = 0.0f;
        for (int w = 0; w < 32; ++w) { SL += redA[w]; SdL += redB[w]; }
        float L_ = 4.0f * zs * trapz_ext(SL, ends[0], ends[2], ends[4]) * (1.0f / PI_F);
        float d_ =             trapz_ext(SdL, ends[1], ends[3], ends[5]) * (1.0f / PI_F);
        __syncthreads();                               // protect LDS reuse
        Lout = L_; dLout = d_;
    };

    auto block_V = [&](float zs, float coef) -> float {
        float fs = eval_f(p, zs);
        float ec = 0.0f, ed = 0.0f;
        if (tid < MQ) { ec = point_Vc(p, zs, fs, tid); ed = point_Vd(p, zs, tid); }
        if      (tid == 0)      { ends[0] = ec; ends[1] = ed; }
        else if (tid == 1)      { ends[2] = ec; }
        else if (tid == MQ - 1) { ends[4] = ec; ends[5] = ed; }
        float sc = wave_sum(ec), sd = wave_sum(ed);
        if (lane == 0) { redA[wid] = sc; redB[wid] = sd; }
        __syncthreads();
        float Sc = 0.0f, Sd = 0.0f;
        for (int w = 0; w < 32; ++w) { Sc += redA[w]; Sd += redB[w]; }
        float Tc = trapz_ext(Sc, ends[0], ends[2], ends[4]);
        float Vc = coef * PI_F * 4.0f * Tc * frcp(zs);
        float Td = 0.5f * (1.0f + ends[1]) * Y0F + H2F * (Sd - 0.5f * (ends[1] + ends[5]));
        float Vd = coef * PI_F * 2.0f * (1.0f - zs) * Td;
        __syncthreads();
        return Vc - Vd;
    };

    float lo = 0.001f, hi = 0.999f, Lv, dLv;
    for (int it = 0; it < 30; ++it) {
        float mid = 0.5f * (lo + hi);
        block_LdL(mid, Lv, dLv);                       // block-uniform result
        if (dLv < 0.0f) hi = mid; else lo = mid;
    }
    float zs_max = 0.5f * (lo + hi);
    block_LdL(zs_max, Lv, dLv);
    float L_max = Lv;

    float coef = expf(logcoef[0]);
    lo = 0.001f; hi = zs_max;
    for (int it = 0; it < 30; ++it) {
        float mid = 0.5f * (lo + hi);
        float v = -block_V(mid, coef);
        if (v < 0.0f) hi = mid; else lo = mid;
    }
    float zs_crit = 0.5f * (lo + hi);
    block_LdL(zs_crit, Lv, dLv);
    float L_crit = Lv;

    if (tid == 0) {
        ws[0] = zs_max; ws[1] = L_max; ws[2] = zs_crit; ws[3] = L_crit;
    }
}

// ---------------------------------------------------------------------------
// Kernel 2: one wave per Ls: 25 fused Newton steps + final integrate_V
// ---------------------------------------------------------------------------
__global__ __launch_bounds__(256)
void adsbh_solve(const float* __restrict__ Ls, const float* __restrict__ a,
                 const float* __restrict__ b, const float* __restrict__ logcoef,
                 const float* __restrict__ ws, float* __restrict__ out, int n) {
    int lane = threadIdx.x & 31;
    int gw   = blockIdx.x * (blockDim.x >> 5) + (threadIdx.x >> 5);
    if (gw >= n) return;                               // wave-uniform

    Params p = make_params(a, b);
    float zs_max = ws[0], L_max = ws[1], L_crit = ws[3];

    float L = Ls[gw];
    bool  valid = L < L_crit;
    float L_eff = valid ? L : 0.5f * L_crit;
    float zs = fminf(fmaxf(L_eff / L_max * zs_max, 1e-4f), 0.9995f);

    for (int it = 0; it < 25; ++it) {
        float Lz, dL;
        wave_L_dL(p, zs, lane, Lz, dL);
        zs = fminf(fmaxf(zs - (Lz - L_eff) / dL, 1e-4f), 0.9995f);
    }

    float coef = expf(logcoef[0]);
    float V = wave_V(p, zs, coef, lane);
    if (lane == 0) out[gw] = valid ? V : 0.0f;
}

// ---------------------------------------------------------------------------
extern "C" void kernel_launch(void* const* d_in, const int* in_sizes, int n_in,
                              void* d_out, int out_size, void* d_ws, size_t ws_size,
                              hipStream_t stream) {
    const float* Ls = (const float*)d_in[0];
    const float* a  = (const float*)d_in[1];
    const float* b  = (const float*)d_in[2];
    const float* lc = (const float*)d_in[3];
    float* out = (float*)d_out;
    float* ws  = (float*)d_ws;
    int n = in_sizes[0];

    adsbh_precompute<<<1, 1024, 0, stream>>>(a, b, lc, ws);

    const int wavesPerBlock = 8;                       // 256 threads = 8 wave32
    int blocks = (n + wavesPerBlock - 1) / wavesPerBlock;
    adsbh_solve<<<blocks, wavesPerBlock * 32, 0, stream>>>(Ls, a, b, lc, ws, out, n);
}